// FrameSimilarity_41351945126515
// MI455X (gfx1250) — compile-verified
//
#include <hip/hip_runtime.h>
#include <hip/hip_bf16.h>

typedef __attribute__((ext_vector_type(2))) float v2f;
typedef __attribute__((ext_vector_type(8))) float v8f;

#define T_LEN   2048
#define CH      768
#define D_DIM   128
#define LOOKUP  101
#define PAD     50
#define BSTRIDE 104   // band row stride, zero-padded so K is a multiple of 4

__device__ __forceinline__ v8f wmma4(v2f a, v2f b, v8f c) {
    // D = A(16x4 f32) x B(4x16 f32) + C(16x16 f32)
    return __builtin_amdgcn_wmma_f32_16x16x4_f32(
        false, a, false, b, (short)0, c, false, false);
}

// ---------------------------------------------------------------------------
// Kernel 1: 7x7 spatial mean.  src (B,C,T,7,7) -> feats (B,768,T) at c_off.
// One block per (b, c, 256-wide t-tile).  Streaming, HBM-bound: use the CDNA5
// async global->LDS pipe (no VGPR staging, no ds_store issue slots; tracked
// by ASYNCcnt) to keep the 23.3 TB/s feed cheap in instructions.
// ---------------------------------------------------------------------------
__global__ void reduce_mean_kernel(const float* __restrict__ src,
                                   float* __restrict__ feats,
                                   int C, int c_off) {
    __shared__ float4 lds4[3136];           // 256 rows * 49 floats = 50176 B
    float* lds = (float*)lds4;

    int blk  = blockIdx.x;
    int tile = blk & 7;                     // 2048 / 256 = 8 tiles per (b,c)
    int bc   = blk >> 3;                    // b*C + c
    int b    = bc / C;
    int c    = bc - b * C;

    const float4* s4 =
        (const float4*)(src + ((size_t)bc * T_LEN + (size_t)tile * 256) * 49);

    // async DMA: global -> LDS, 16 B per lane per issue
    for (int i = threadIdx.x; i < 3136; i += 256) {
        unsigned int   loff = (unsigned int)(uintptr_t)(lds4 + i);  // LDS byte offset
        const float4*  gptr = s4 + i;
        asm volatile("global_load_async_to_lds_b128 %0, %1, off"
                     :: "v"(loff), "v"(gptr) : "memory");
    }
    asm volatile("s_wait_asynccnt 0" ::: "memory");   // this wave's DMAs done
    __syncthreads();                                  // visible to all waves

    float s = 0.f;
    int base = threadIdx.x * 49;            // stride 49: conflict-free (gcd(49,64)=1)
#pragma unroll
    for (int k = 0; k < 49; ++k) s += lds[base + k];

    feats[((size_t)(b * CH + c_off + c)) * T_LEN + tile * 256 + threadIdx.x] =
        s * (1.0f / 49.0f);
}

// ---------------------------------------------------------------------------
// Kernel 2: x = feats^T @ Wp  (per batch: T x 768 * 768 x 128), then L2-norm
// rows.  Block = 256 thr = 8 waves; wave w computes d-tile w of one t-tile.
// ---------------------------------------------------------------------------
__global__ void project_norm_kernel(const float* __restrict__ feats,
                                    const float* __restrict__ Wp,
                                    float* __restrict__ xn) {
    __shared__ float xs[16][D_DIM];
    __shared__ float inv[16];

    int b    = blockIdx.x >> 7;             // 128 t-tiles per batch
    int t0   = (blockIdx.x & 127) << 4;
    int lane = threadIdx.x & 31;
    int w    = threadIdx.x >> 5;            // d-tile
    int mn   = lane & 15;
    int kb   = (lane >> 4) << 1;            // 0 for lanes 0-15, 2 for 16-31
    int d0   = w << 4;

    const float* F = feats + (size_t)b * CH * T_LEN;
    v8f acc = {};
#pragma unroll 4
    for (int k = 0; k < CH; k += 4) {
        v2f a, bf;
        a.x  = F[(size_t)(k + kb)     * T_LEN + t0 + mn];
        a.y  = F[(size_t)(k + kb + 1) * T_LEN + t0 + mn];
        bf.x = Wp[(k + kb)     * D_DIM + d0 + mn];
        bf.y = Wp[(k + kb + 1) * D_DIM + d0 + mn];
        acc = wmma4(a, bf, acc);
    }

    int mhi = (lane >> 4) << 3;             // +8 rows for upper half-wave
#pragma unroll
    for (int i = 0; i < 8; ++i)
        xs[i + mhi][d0 + mn] = acc[i];
    __syncthreads();

    if (threadIdx.x < 16) {
        float s = 0.f;
#pragma unroll
        for (int c = 0; c < D_DIM; ++c) s += xs[threadIdx.x][c] * xs[threadIdx.x][c];
        float nrm = sqrtf(s);
        inv[threadIdx.x] = 1.0f / fmaxf(nrm, 1e-12f);
    }
    __syncthreads();

    int r = threadIdx.x >> 4;               // 16 threads per 128-float row
    int c0 = (threadIdx.x & 15) << 3;
    float iv = inv[r];
    float* dst = xn + ((size_t)(b * T_LEN + t0 + r)) * D_DIM + c0;
#pragma unroll
    for (int j = 0; j < 8; ++j) dst[j] = xs[r][c0 + j] * iv;
}

// ---------------------------------------------------------------------------
// Kernel 3: banded self-similarity. band[b][t][s] = xn[t] . xn[t+s-50],
// s in [0,101); rows padded to 104 and zero-initialized.  One wave per
// 16-row t-tile; 9 u-tiles of 16x16 sims via 32 WMMA K-steps each.
// ---------------------------------------------------------------------------
__global__ void band_sims_kernel(const float* __restrict__ xn,
                                 float* __restrict__ band) {
    int b    = blockIdx.x >> 7;
    int t0   = (blockIdx.x & 127) << 4;
    int lane = threadIdx.x & 31;
    int mn   = lane & 15;
    int kb   = (lane >> 4) << 1;
    int mhi  = (lane >> 4) << 3;

    const float* X  = xn   + (size_t)b * T_LEN * D_DIM;
    float*       Bd = band + (size_t)b * T_LEN * BSTRIDE;

    // zero this tile's band rows (covers edge entries never scattered;
    // same-wave stores are ordered, so later scatters win)
    for (int i = lane; i < 16 * BSTRIDE; i += 32) {
        int r = i / BSTRIDE;
        int s = i - r * BSTRIDE;
        Bd[(size_t)(t0 + r) * BSTRIDE + s] = 0.f;
    }

    // preload all A fragments (this tile's rows), K = 128 -> 32 steps
    v2f aF[32];
#pragma unroll
    for (int k4 = 0; k4 < 32; ++k4) {
        int k = (k4 << 2) + kb;
        const float* p = &X[(size_t)(t0 + mn) * D_DIM + k];
        aF[k4].x = p[0];
        aF[k4].y = p[1];
    }

    for (int ut = -4; ut <= 4; ++ut) {
        int u0 = t0 + ut * 16;
        if (u0 + 15 < 0 || u0 >= T_LEN) continue;   // wave-uniform skip

        int  urow = u0 + mn;
        bool ok   = (urow >= 0) && (urow < T_LEN);
        const float* up = &X[(size_t)(ok ? urow : 0) * D_DIM];

        v8f acc = {};
#pragma unroll
        for (int k4 = 0; k4 < 32; ++k4) {
            int k = (k4 << 2) + kb;
            v2f bf;
            bf.x = ok ? up[k]     : 0.f;
            bf.y = ok ? up[k + 1] : 0.f;
            acc = wmma4(aF[k4], bf, acc);
        }

        // scatter D(16x16) into the band:  s = u - t + 50
#pragma unroll
        for (int i = 0; i < 8; ++i) {
            int t = t0 + i + mhi;
            int u = u0 + mn;
            int s = u - t + PAD;
            if (u >= 0 && u < T_LEN && s >= 0 && s < LOOKUP)
                Bd[(size_t)t * BSTRIDE + s] = acc[i];
        }
    }
}

// ---------------------------------------------------------------------------
// Kernel 4: out = relu(band @ fc_w + fc_b).  K = 104 (rows >= 101 are zero on
// both sides).  Block = 8 waves; wave w -> output d-tile w of one t-tile.
// ---------------------------------------------------------------------------
__global__ void fc_relu_kernel(const float* __restrict__ band,
                               const float* __restrict__ fcw,
                               const float* __restrict__ fcb,
                               float* __restrict__ out) {
    int b    = blockIdx.x >> 7;
    int t0   = (blockIdx.x & 127) << 4;
    int lane = threadIdx.x & 31;
    int w    = threadIdx.x >> 5;
    int mn   = lane & 15;
    int kb   = (lane >> 4) << 1;
    int mhi  = (lane >> 4) << 3;
    int j0   = w << 4;

    const float* Bd = band + (size_t)b * T_LEN * BSTRIDE;

    v8f acc = {};
#pragma unroll
    for (int k4 = 0; k4 < 26; ++k4) {
        int k = (k4 << 2) + kb;
        v2f a, bf;
        const float* p = &Bd[(size_t)(t0 + mn) * BSTRIDE + k];
        a.x  = p[0];
        a.y  = p[1];
        bf.x = (k     < LOOKUP) ? fcw[k       * D_DIM + j0 + mn] : 0.f;
        bf.y = (k + 1 < LOOKUP) ? fcw[(k + 1) * D_DIM + j0 + mn] : 0.f;
        acc = wmma4(a, bf, acc);
    }

    float bias = fcb[j0 + mn];
#pragma unroll
    for (int i = 0; i < 8; ++i) {
        int t = t0 + i + mhi;
        out[(size_t)(b * T_LEN + t) * D_DIM + j0 + mn] = fmaxf(acc[i] + bias, 0.f);
    }
}

// ---------------------------------------------------------------------------
extern "C" void kernel_launch(void* const* d_in, const int* in_sizes, int n_in,
                              void* d_out, int out_size, void* d_ws, size_t ws_size,
                              hipStream_t stream) {
    const float* x0  = (const float*)d_in[0];   // (2,256,2048,7,7)
    const float* x1  = (const float*)d_in[1];   // (2,512,2048,7,7)
    const float* Wp  = (const float*)d_in[2];   // (768,128)
    const float* fcw = (const float*)d_in[3];   // (101,128)
    const float* fcb = (const float*)d_in[4];   // (128,)
    float*       out = (float*)d_out;           // (2,2048,128)

    float* feats = (float*)d_ws;                          // 2*768*2048
    float* xn    = feats + (size_t)2 * CH * T_LEN;        // 2*2048*128
    float* band  = xn    + (size_t)2 * T_LEN * D_DIM;     // 2*2048*104

    // 1) spatial means (HBM-bound, ~616 MB streamed via async global->LDS)
    reduce_mean_kernel<<<2 * 256 * 8, 256, 0, stream>>>(x0, feats, 256, 0);
    reduce_mean_kernel<<<2 * 512 * 8, 256, 0, stream>>>(x1, feats, 512, 256);

    // 2) projection + L2 normalize
    project_norm_kernel<<<2 * 128, 256, 0, stream>>>(feats, Wp, xn);

    // 3) banded similarities (only the +/-50 band, never full sims)
    band_sims_kernel<<<2 * 128, 32, 0, stream>>>(xn, band);

    // 4) band @ fc_w + bias, relu
    fc_relu_kernel<<<2 * 128, 256, 0, stream>>>(band, fcw, fcb, out);
}